// MockRouter_76192719831297
// MI455X (gfx1250) — compile-verified
//
#include <hip/hip_runtime.h>

typedef __attribute__((ext_vector_type(2))) float v2f;
typedef __attribute__((ext_vector_type(8))) float v8f;

#define DDIM   4096
#define NEXP   64
#define TOPK   8
#define KHALF  2048
#define KCHUNK 64
#define NCHUNK (KHALF / KCHUNK)   // 32
#define XS     68   // padded LDS stride for 64-float rows (bank-conflict free)
#define OS     65   // padded stride for logits scratch

// Per-buffer layout (floats): lds_x [2 halves][32 tok][XS] = 4352, then
// lds_w [2 halves][64 exp][XS] = 8704  -> BUFSZ = 13056 floats.
// Two buffers = 26112 floats = 104448 B. Logits scratch (4160 floats)
// aliases buffer 0's x-region after the final barrier.
#define BUFSZ 13056
#define WOFF  4352

// Async copy global -> LDS (CDNA5 GLOBAL_LOAD_ASYNC_TO_LDS_B128, ASYNCcnt).
// No data VGPRs: transfers stay in flight until s_wait_asynccnt.
__device__ __forceinline__ void async_copy_b128(unsigned lds_off, const float* gptr) {
    asm volatile("global_load_async_to_lds_b128 %0, %1, off"
                 :: "v"(lds_off), "v"((unsigned long long)(uintptr_t)gptr)
                 : "memory");
}

__global__ __launch_bounds__(128)
void router_wmma_kernel(const float* __restrict__ x,
                        const float* __restrict__ gw,
                        float* __restrict__ out_w,
                        int*   __restrict__ out_i)
{
    __shared__ __align__(16) float smem[2 * BUFSZ];
    float* lds_p = smem;   // logits scratch, reuses buffer 0 after final barrier

    const int t    = threadIdx.x;
    const int wave = t >> 5;
    const int lane = t & 31;
    const int l15  = lane & 15;
    const int lhi  = lane >> 4;               // 0 or 1
    const int tok0 = blockIdx.x * 32;

    const int h  = wave & 1;                  // K-half owned by this wave
    const int tt = wave >> 1;                 // token row-tile (0 or 1)

    v8f acc[4] = {};                          // 16 tok x 64 experts, 4 n-tiles

    // LDS read bases (buffer 0); add BUFSZ for buffer 1
    const float* a_base = &smem[h * 32 * XS + (tt * 16 + l15) * XS + lhi * 2];
    const float* b_base = &smem[WOFF + h * 64 * XS + l15 * XS + lhi * 2];

    // Issue one chunk's 24 async b128 copies (x-tile: 8, w-tile: 16) into buffer `buf`
    auto stage = [&](int buf, int kcol) {
        float* bx = smem + buf * BUFSZ;
        float* bw = bx + WOFF;
        #pragma unroll
        for (int i = 0; i < 8; ++i) {          // x: 2 halves x 32 rows x 16 float4
            int idx = t + i * 128;
            int hh  = idx >> 9;
            int rem = idx & 511;
            int row = rem >> 4;
            int c4  = rem & 15;
            const float* gp = &x[(size_t)(tok0 + row) * DDIM + hh * KHALF + kcol + c4 * 4];
            async_copy_b128((unsigned)(uintptr_t)&bx[hh * 32 * XS + row * XS + c4 * 4], gp);
        }
        #pragma unroll
        for (int i = 0; i < 16; ++i) {         // w: 2 halves x 64 experts x 16 float4
            int idx = t + i * 128;
            int hh  = idx >> 10;
            int rem = idx & 1023;
            int row = rem >> 4;
            int c4  = rem & 15;
            const float* gp = &gw[(size_t)row * DDIM + hh * KHALF + kcol + c4 * 4];
            async_copy_b128((unsigned)(uintptr_t)&bw[hh * 64 * XS + row * XS + c4 * 4], gp);
        }
    };

    stage(0, 0);                               // prologue: fill buffer 0

    for (int chunk = 0; chunk < NCHUNK; ++chunk) {
        if (chunk + 1 < NCHUNK) {
            // prefetch next chunk into the other buffer, THEN wait for current:
            // async completions are in-order, so <=24 outstanding means the
            // previous chunk's 24 transfers have all landed in LDS.
            stage((chunk + 1) & 1, (chunk + 1) * KCHUNK);
            asm volatile("s_wait_asynccnt 0x18" ::: "memory");   // 24
        } else {
            asm volatile("s_wait_asynccnt 0x0" ::: "memory");
        }
        __syncthreads();

        const float* ab = a_base + (chunk & 1) * BUFSZ;
        const float* bb = b_base + (chunk & 1) * BUFSZ;

        // ---- 16 K-steps of 4: A frag + 4 B frags + 4 f32 WMMAs ----
        #pragma unroll
        for (int kk = 0; kk < KCHUNK; kk += 4) {
            v2f a  = *(const v2f*)&ab[kk];
            v2f b0 = *(const v2f*)&bb[kk];                // experts  0..15
            v2f b1 = *(const v2f*)&bb[16 * XS + kk];      // experts 16..31
            v2f b2 = *(const v2f*)&bb[32 * XS + kk];      // experts 32..47
            v2f b3 = *(const v2f*)&bb[48 * XS + kk];      // experts 48..63
            acc[0] = __builtin_amdgcn_wmma_f32_16x16x4_f32(false, a, false, b0, (short)0, acc[0], false, false);
            acc[1] = __builtin_amdgcn_wmma_f32_16x16x4_f32(false, a, false, b1, (short)0, acc[1], false, false);
            acc[2] = __builtin_amdgcn_wmma_f32_16x16x4_f32(false, a, false, b2, (short)0, acc[2], false, false);
            acc[3] = __builtin_amdgcn_wmma_f32_16x16x4_f32(false, a, false, b3, (short)0, acc[3], false, false);
        }
        __syncthreads();   // buffer free for the chunk+2 stage
    }

    // ---- dump per-wave (K-half) partial logits: C layout VGPR v -> M = v + 8*lhi ----
    #pragma unroll
    for (int nt = 0; nt < 4; ++nt) {
        #pragma unroll
        for (int v = 0; v < 8; ++v) {
            lds_p[wave * 16 * OS + (v + lhi * 8) * OS + nt * 16 + l15] = acc[nt][v];
        }
    }
    __syncthreads();

    // ---- per-token epilogue: sum K-halves, top-8, sigmoid, normalize ----
    if (t < 32) {
        const int pt = (t >> 4) * 2;           // partial pair: waves {0,1} or {2,3}
        const int r  = (t & 15) * OS;
        float tv[TOPK];
        int   ti[TOPK];
        #pragma unroll
        for (int j = 0; j < TOPK; ++j) { tv[j] = -3.4e38f; ti[j] = 0; }

        for (int e = 0; e < NEXP; ++e) {
            float v = lds_p[pt * 16 * OS + r + e] + lds_p[(pt + 1) * 16 * OS + r + e];
            if (v > tv[TOPK - 1]) {            // strict > keeps lowest-index-first on ties
                int j = TOPK - 1;
                while (j > 0 && v > tv[j - 1]) {
                    tv[j] = tv[j - 1]; ti[j] = ti[j - 1]; --j;
                }
                tv[j] = v; ti[j] = e;
            }
        }

        float w8[TOPK];
        float s = 0.0f;
        #pragma unroll
        for (int j = 0; j < TOPK; ++j) {
            w8[j] = 1.0f / (1.0f + expf(-tv[j]));   // sigmoid of gathered logits
            s += w8[j];
        }
        s = fmaxf(s, 1e-12f);
        const float inv = 1.0f / s;
        const int tok = tok0 + t;
        #pragma unroll
        for (int j = 0; j < TOPK; ++j) {
            out_w[tok * TOPK + j] = w8[j] * inv;
            out_i[tok * TOPK + j] = ti[j];
        }
    }
}

extern "C" void kernel_launch(void* const* d_in, const int* in_sizes, int n_in,
                              void* d_out, int out_size, void* d_ws, size_t ws_size,
                              hipStream_t stream) {
    (void)in_sizes; (void)n_in; (void)out_size; (void)d_ws; (void)ws_size;
    const float* x  = (const float*)d_in[0];   // (8192, 4096) f32
    const float* gw = (const float*)d_in[1];   // (64, 4096) f32
    float* out_w = (float*)d_out;                          // weights (8192,8) f32
    int*   out_i = (int*)((float*)d_out + 8192 * TOPK);    // indices (8192,8) i32

    dim3 grid(8192 / 32);
    dim3 block(128);
    router_wmma_kernel<<<grid, block, 0, stream>>>(x, gw, out_w, out_i);
}